// MatchingNetwork_38182259261917
// MI455X (gfx1250) — compile-verified
//
#include <hip/hip_runtime.h>
#include <math.h>

typedef __attribute__((ext_vector_type(2))) float v2f;
typedef __attribute__((ext_vector_type(8))) float v8f;

#define DIM   256
#define NSUP  320
#define NQ    2048
#define NWAY  20

// ---------------------------------------------------------------------------
// GEMM: out[M x DIM] = A[M x DIM] @ B[DIM x DIM] (+ bias broadcast over rows)
// One wave per 16x16 output tile, V_WMMA_F32_16X16X4_F32, K-loop step 4.
//
// A-fragment layout (ISA 7.12.2, 32-bit A 16x4):
//   lanes 0-15  : VGPR0 = A[M=lane][K=k+0], VGPR1 = A[M=lane][K=k+1]
//   lanes 16-31 : VGPR0 = A[M=lane-16][K=k+2], VGPR1 = A[..][K=k+3]
// B-fragment (4x16, rows striped across lanes mirroring A's K split):
//   lanes 0-15  : VGPR0 = B[K=k+0][N=lane], VGPR1 = B[K=k+1][N=lane]
//   lanes 16-31 : VGPR0 = B[K=k+2][N=lane-16], VGPR1 = B[K=k+3][N=lane-16]
// C/D layout: VGPR i -> (M = tm + i + 8*half, N = tn + lane&15)
// ---------------------------------------------------------------------------
__global__ __launch_bounds__(256) void gemm_wmma_f32(
    const float* __restrict__ A, const float* __restrict__ B,
    const float* __restrict__ bias, float* __restrict__ out, int M)
{
    const int lane  = threadIdx.x & 31;
    const int wave  = threadIdx.x >> 5;
    const int tilesN = DIM / 16;                 // 16
    const int tile   = blockIdx.x * 8 + wave;    // one 16x16 tile per wave
    const int nTiles = (M / 16) * tilesN;
    if (tile >= nTiles) return;                  // grid sized exactly; uniform per wave

    const int tm   = (tile / tilesN) * 16;
    const int tn   = (tile % tilesN) * 16;
    const int half = lane >> 4;                  // 0 or 1 (K sub-pair)
    const int l16  = lane & 15;

    const float* __restrict__ Arow = A + (size_t)(tm + l16) * DIM;
    const float* __restrict__ Bcol = B + tn + l16;

    v8f acc = {};
    #pragma unroll 4
    for (int k = 0; k < DIM; k += 4) {
        const int ka = k + half * 2;
        v2f a;                                   // contiguous pair -> b64 load
        a.x = Arow[ka + 0];
        a.y = Arow[ka + 1];
        v2f b;
        b.x = Bcol[(size_t)(ka + 0) * DIM];
        b.y = Bcol[(size_t)(ka + 1) * DIM];
        acc = __builtin_amdgcn_wmma_f32_16x16x4_f32(
            /*neg_a=*/false, a, /*neg_b=*/false, b,
            /*c_mod=*/(short)0, acc, /*reuse_a=*/false, /*reuse_b=*/false);
    }

    const float bv = bias ? bias[tn + l16] : 0.0f;
    #pragma unroll
    for (int i = 0; i < 8; ++i)
        out[(size_t)(tm + i + half * 8) * DIM + tn + l16] = acc[i] + bv;
}

// ---------------------------------------------------------------------------
// Fused scoring + softmax + one-hot aggregation. One block (8 waves) per query.
//   score[s] = sum_d relu(qp[q,d] + spb[s,d]) * W2[d] + b2
//   logits[q,c] = sum_{s: label[s]=c} exp(score[s]-max) / sum_s exp(score[s]-max)
// Lane owns d = lane*8 .. lane*8+7 (query row + W2 cached in registers);
// waves stripe the 320 support rows; spb (320 KB) is L2/WGP$-resident.
// ---------------------------------------------------------------------------
__global__ __launch_bounds__(256) void score_softmax_kernel(
    const float* __restrict__ qp, const float* __restrict__ spb,
    const float* __restrict__ W2, const float* __restrict__ b2,
    const int* __restrict__ labels, float* __restrict__ out)
{
    __shared__ float s_scores[NSUP];
    __shared__ float s_red[8];
    __shared__ float s_bins[NWAY];
    __shared__ float s_stat[2];                  // [0]=max, [1]=sum

    const int q    = blockIdx.x;
    const int tid  = threadIdx.x;
    const int lane = tid & 31;
    const int wave = tid >> 5;

    const float4 q0 = *(const float4*)(qp + (size_t)q * DIM + lane * 8);
    const float4 q1 = *(const float4*)(qp + (size_t)q * DIM + lane * 8 + 4);
    const float4 w0 = *(const float4*)(W2 + lane * 8);
    const float4 w1 = *(const float4*)(W2 + lane * 8 + 4);
    const float  b2v = b2[0];

    for (int s = wave; s < NSUP; s += 8) {
        const float* __restrict__ sp = spb + (size_t)s * DIM + lane * 8;
        const float4 p0 = *(const float4*)(sp);
        const float4 p1 = *(const float4*)(sp + 4);
        float acc;
        acc  = fmaxf(q0.x + p0.x, 0.0f) * w0.x;
        acc += fmaxf(q0.y + p0.y, 0.0f) * w0.y;
        acc += fmaxf(q0.z + p0.z, 0.0f) * w0.z;
        acc += fmaxf(q0.w + p0.w, 0.0f) * w0.w;
        acc += fmaxf(q1.x + p1.x, 0.0f) * w1.x;
        acc += fmaxf(q1.y + p1.y, 0.0f) * w1.y;
        acc += fmaxf(q1.z + p1.z, 0.0f) * w1.z;
        acc += fmaxf(q1.w + p1.w, 0.0f) * w1.w;
        #pragma unroll
        for (int off = 16; off; off >>= 1)
            acc += __shfl_xor(acc, off, 32);
        if (lane == 0) s_scores[s] = acc + b2v;
    }
    __syncthreads();

    // ---- block max over 320 scores ----
    float m = -INFINITY;
    for (int i = tid; i < NSUP; i += 256) m = fmaxf(m, s_scores[i]);
    #pragma unroll
    for (int off = 16; off; off >>= 1)
        m = fmaxf(m, __shfl_xor(m, off, 32));
    if (lane == 0) s_red[wave] = m;
    __syncthreads();
    if (tid == 0) {
        float mm = s_red[0];
        #pragma unroll
        for (int i = 1; i < 8; ++i) mm = fmaxf(mm, s_red[i]);
        s_stat[0] = mm;
    }
    if (tid < NWAY) s_bins[tid] = 0.0f;
    __syncthreads();
    const float mx = s_stat[0];

    // ---- exp + per-class accumulation (fused softmax @ one-hot) ----
    float psum = 0.0f;
    for (int i = tid; i < NSUP; i += 256) {
        const float e = __expf(s_scores[i] - mx);
        psum += e;
        atomicAdd(&s_bins[labels[i]], e);        // ds_add_f32
    }
    #pragma unroll
    for (int off = 16; off; off >>= 1)
        psum += __shfl_xor(psum, off, 32);
    if (lane == 0) s_red[wave] = psum;
    __syncthreads();
    if (tid == 0) {
        float ss = 0.0f;
        #pragma unroll
        for (int i = 0; i < 8; ++i) ss += s_red[i];
        s_stat[1] = ss;
    }
    __syncthreads();

    const float inv = 1.0f / s_stat[1];
    if (tid < NWAY) out[(size_t)q * NWAY + tid] = s_bins[tid] * inv;
}

// ---------------------------------------------------------------------------
extern "C" void kernel_launch(void* const* d_in, const int* in_sizes, int n_in,
                              void* d_out, int out_size, void* d_ws, size_t ws_size,
                              hipStream_t stream) {
    const float* support = (const float*)d_in[0];   // [320, 256]
    const float* query   = (const float*)d_in[1];   // [2048, 256]
    const int*   labels  = (const int*)  d_in[2];   // [320] int32
    const float* W1      = (const float*)d_in[3];   // [512, 256]
    const float* b1      = (const float*)d_in[4];   // [256]
    const float* W2      = (const float*)d_in[5];   // [256]
    const float* b2      = (const float*)d_in[6];   // [1]

    float* qp  = (float*)d_ws;                      // [2048, 256]  (2 MB)
    float* spb = qp + (size_t)NQ * DIM;             // [320, 256]   (320 KB)

    // q_proj = query @ W1[:256]          -> 2048 tiles / 8 waves = 256 blocks
    gemm_wmma_f32<<<256, 256, 0, stream>>>(query, W1, nullptr, qp, NQ);
    // s_proj = support @ W1[256:] + b1   -> 320 tiles / 8 waves = 40 blocks
    gemm_wmma_f32<<<40, 256, 0, stream>>>(support, W1 + (size_t)DIM * DIM, b1, spb, NSUP);
    // fused scores -> softmax -> class bins
    score_softmax_kernel<<<NQ, 256, 0, stream>>>(qp, spb, W2, b2, labels, (float*)d_out);
}